// FingerprintAttention_18013092839568
// MI455X (gfx1250) — compile-verified
//
#include <hip/hip_runtime.h>
#include <math.h>

#define EMB 128
#define CHUNK 24                          // tokens per pipeline chunk (kernel 2)
#define F4_PER_THREAD (CHUNK * 32 / 256)  // float4 copies per thread per array = 3
#define ASYNC_PER_CHUNK (2 * F4_PER_THREAD)

#define ROWS_PER_BLOCK 64                 // kernel 1 tile rows
#define LDS_STRIDE 132                    // 128 + 4 pad -> conflict-free column gathers

typedef float v2f __attribute__((ext_vector_type(2)));
typedef float v8f __attribute__((ext_vector_type(8)));

// ---------- CDNA5 async global->LDS copy (ASYNCcnt-tracked) ----------
__device__ __forceinline__ void async_copy_f4(unsigned lds_byte_addr,
                                              unsigned gbyte_off,
                                              unsigned long long base) {
  // GVS mode: mem_addr = SADDR(64) + VADDR(32); vdst VGPR = LDS byte address
  asm volatile("global_load_async_to_lds_b128 %0, %1, %2"
               :: "v"(lds_byte_addr), "v"(gbyte_off), "s"(base)
               : "memory");
}

template <int N>
__device__ __forceinline__ void wait_asynccnt() {
  asm volatile("s_wait_asynccnt %0" :: "i"(N) : "memory");
}

// ---------- Kernel 1: scores via chained V_WMMA_F32_16X16X4_F32 ----------
// scores[i] = dot(key[i,:], diag(w)*fp) / sqrt(E)
// A = all-ones 16x4 (layout-invariant), B[k,n] = key[row n, k]*dwf[k].
// D[m,n] accumulates score of tile row n over 32 K-slabs of 4.
__global__ __launch_bounds__(128) void fa_scores_wmma_kernel(
    const float* __restrict__ key, const float* __restrict__ fp,
    const float* __restrict__ w, float* __restrict__ scores, int T) {
  __shared__ float diag[EMB];
  __shared__ __align__(16) float kd[ROWS_PER_BLOCK * LDS_STRIDE];

  const int tid = threadIdx.x;  // 128 threads
  diag[tid] = w[tid * EMB + tid] * fp[tid];
  __syncthreads();

  // Stage 64 rows of key, pre-multiplied by dwf, coalesced float4 loads.
  const size_t rowbase_blk = (size_t)blockIdx.x * ROWS_PER_BLOCK;
  const float4* key4 = reinterpret_cast<const float4*>(key);
  const float4* d4 = reinterpret_cast<const float4*>(diag);
  const unsigned maxq = (unsigned)((size_t)T * EMB / 4 - 1);  // last valid f4
  #pragma unroll
  for (int u = 0; u < (ROWS_PER_BLOCK * EMB / 4) / 128; ++u) {  // 16 iters
    const unsigned q = (unsigned)tid + 128u * (unsigned)u;  // f4 idx in tile
    unsigned gq = (unsigned)(rowbase_blk * (EMB / 4)) + q;
    if (gq > maxq) gq = maxq;  // clamp tail; stores are guarded below
    const float4 kk = key4[gq];
    const float4 dd = d4[q & 31];           // column group = q % 32
    const int row = (int)(q >> 5);
    const int colf4 = (int)(q & 31);
    float4* dst = reinterpret_cast<float4*>(&kd[row * LDS_STRIDE + colf4 * 4]);
    *dst = make_float4(kk.x * dd.x, kk.y * dd.y, kk.z * dd.z, kk.w * dd.w);
  }
  __syncthreads();

  // Each wave (4 waves) owns 16 rows; 32 chained WMMAs over K.
  const int wv = tid >> 5;
  const int lane = tid & 31;
  const int n = lane & 15;                 // B/D column = tile row index
  const int khalf = (lane >> 4) << 1;      // 0 or 2 (K slot within slab)
  const float* rowptr = &kd[(wv * 16 + n) * LDS_STRIDE];

  v8f c = {0.f, 0.f, 0.f, 0.f, 0.f, 0.f, 0.f, 0.f};
  v2f a; a.x = 1.0f; a.y = 1.0f;           // all-ones A: slot-mapping invariant
  #pragma unroll
  for (int s = 0; s < 32; ++s) {
    const int kb = s * 4 + khalf;
    v2f b; b.x = rowptr[kb]; b.y = rowptr[kb + 1];
    c = __builtin_amdgcn_wmma_f32_16x16x4_f32(
        /*neg_a=*/false, a, /*neg_b=*/false, b,
        /*c_mod=*/(short)0, c, /*reuse_a=*/false, /*reuse_b=*/false);
  }

  // D layout: VGPR0, lanes 0-15 -> (M=0, N=lane). Lanes 0-15 store 16 scores.
  const size_t row = rowbase_blk + (size_t)(wv * 16 + n);
  if (lane < 16 && row < (size_t)T)
    scores[row] = c[0] * 0.08838834764831845f;  // 1/sqrt(128)
}

// ---------- Kernel 2: per-segment softmax + weighted feature sum ----------
__global__ __launch_bounds__(256) void fa_segment_kernel(
    const float* __restrict__ value, const float* __restrict__ key,
    const int* __restrict__ seg, const float* __restrict__ scores,
    float* __restrict__ out, int T) {
  const int b = blockIdx.x;
  const int tid = threadIdx.x;

  __shared__ int sbounds[2];
  __shared__ float sred[8];
  __shared__ float sbcast;
  __shared__ __align__(16) float buf[2][CHUNK * 2 * EMB];  // [val | key] chunks
  __shared__ float swts[CHUNK];

  if (tid == 0) {
    int lo = 0, hi = T;
    while (lo < hi) { int m = (lo + hi) >> 1; if (seg[m] < b) lo = m + 1; else hi = m; }
    sbounds[0] = lo;
    int lo2 = lo; hi = T;
    while (lo2 < hi) { int m = (lo2 + hi) >> 1; if (seg[m] < b + 1) lo2 = m + 1; else hi = m; }
    sbounds[1] = lo2;
  }
  __syncthreads();
  const int start = sbounds[0];
  const int end = sbounds[1];
  const int len = end - start;
  if (len <= 0) { out[(size_t)b * (2 * EMB) + tid] = 0.0f; return; }

  // ---- pass 1: segment max ----
  float m = -3.402823466e38f;
  for (int i = start + tid; i < end; i += 256) m = fmaxf(m, scores[i]);
  #pragma unroll
  for (int off = 16; off > 0; off >>= 1) m = fmaxf(m, __shfl_xor(m, off, 32));
  if ((tid & 31) == 0) sred[tid >> 5] = m;
  __syncthreads();
  if (tid == 0) {
    float mm = sred[0];
    #pragma unroll
    for (int i = 1; i < 8; ++i) mm = fmaxf(mm, sred[i]);
    sbcast = mm;
  }
  __syncthreads();
  const float smax = sbcast;

  // ---- pass 2: segment sum of exp ----
  float ssum = 0.0f;
  for (int i = start + tid; i < end; i += 256) ssum += expf(scores[i] - smax);
  #pragma unroll
  for (int off = 16; off > 0; off >>= 1) ssum += __shfl_xor(ssum, off, 32);
  __syncthreads();  // sred reuse
  if ((tid & 31) == 0) sred[tid >> 5] = ssum;
  __syncthreads();
  if (tid == 0) {
    float t = 0.0f;
    #pragma unroll
    for (int i = 0; i < 8; ++i) t += sred[i];
    sbcast = t;
  }
  __syncthreads();
  const float invden = 1.0f / sbcast;

  // ---- pass 3: double-buffered async-LDS weighted accumulation ----
  const int nchunks = (len + CHUNK - 1) / CHUNK;
  const unsigned long long vbase = (unsigned long long)value;
  const unsigned long long kbase = (unsigned long long)key;
  const unsigned lds0 = (unsigned)(unsigned long long)(void*)&buf[0][0];
  const unsigned maxvoff = (unsigned)((size_t)T * EMB * 4 - 16);  // last valid f4

  auto issue_chunk = [&](int ck, int bsel) {
    const unsigned c0 = (unsigned)((size_t)(start + ck * CHUNK) * EMB * 4);
    const unsigned ldsb = lds0 + (unsigned)bsel * (CHUNK * 2 * EMB * 4);
    #pragma unroll
    for (int u = 0; u < F4_PER_THREAD; ++u) {
      const unsigned q = (unsigned)tid + 256u * (unsigned)u;  // f4 idx in chunk
      unsigned off = c0 + q * 16u;
      if (off > maxvoff) off = maxvoff;  // clamp; masked by zero weight
      async_copy_f4(ldsb + q * 16u, off, vbase);
      async_copy_f4(ldsb + (unsigned)(CHUNK * EMB * 4) + q * 16u, off, kbase);
    }
  };

  issue_chunk(0, 0);
  float acc = 0.0f;
  const int col = tid;  // 0..255: 0..127 -> value cols, 128..255 -> key cols

  for (int ck = 0; ck < nchunks; ++ck) {
    const bool nb = (ck + 1 < nchunks);
    if (nb) issue_chunk(ck + 1, (ck + 1) & 1);
    // in-order async completion: <= ASYNC_PER_CHUNK outstanding means chunk ck landed
    if (nb) wait_asynccnt<ASYNC_PER_CHUNK>();
    else    wait_asynccnt<0>();

    if (tid < CHUNK) {
      const int t = start + ck * CHUNK + tid;
      swts[tid] = (t < end) ? expf(scores[t] - smax) * invden : 0.0f;
    }
    __syncthreads();  // async-LDS visibility across waves + swts ready

    const float* bb = &buf[ck & 1][0];
    const float* src = (col < EMB) ? (bb + col) : (bb + CHUNK * EMB + (col - EMB));
    #pragma unroll
    for (int t = 0; t < CHUNK; ++t) acc += swts[t] * src[(size_t)t * EMB];
    __syncthreads();  // buffer consumed before it is overwritten next round
  }

  out[(size_t)b * (2 * EMB) + col] = acc;
}

extern "C" void kernel_launch(void* const* d_in, const int* in_sizes, int n_in,
                              void* d_out, int out_size, void* d_ws, size_t ws_size,
                              hipStream_t stream) {
  const float* value = (const float*)d_in[0];
  const float* key   = (const float*)d_in[1];
  const int*   seg   = (const int*)d_in[2];
  const float* fp    = (const float*)d_in[3];
  const float* w     = (const float*)d_in[4];
  float* out = (float*)d_out;

  const int T = in_sizes[2];           // number of tokens (segment_ids length)
  const int B = out_size / (2 * EMB);  // number of segments

  float* scores = (float*)d_ws;        // T floats of scratch

  fa_scores_wmma_kernel<<<dim3((T + ROWS_PER_BLOCK - 1) / ROWS_PER_BLOCK),
                          dim3(128), 0, stream>>>(key, fp, w, scores, T);
  fa_segment_kernel<<<dim3(B), dim3(256), 0, stream>>>(value, key, seg, scores, out, T);
}